// _NonLocalBlockND_65833258713357
// MI455X (gfx1250) — compile-verified
//
#include <hip/hip_runtime.h>
#include <math.h>

typedef __attribute__((ext_vector_type(2))) float v2f;
typedef __attribute__((ext_vector_type(8))) float v8f;

#define B_   4
#define C_   256
#define N_   4096      // H*W = 64*64
#define CI_  128

// workspace layout in floats
#define WS_XS   0      // B*C   = 1024
#define WS_W    1024   // B*C   = 1024
#define WS_S    2048   // B     = 4
#define WS_CONF 2064   // B*N   = 16384  (16B aligned: 2064*4 = 8256)

// ---------------- Pass 1: xs[b,c] = sum_n x[b,c,n] ----------------
__global__ void nlb_colsum(const float* __restrict__ x, float* __restrict__ xs) {
    const int bc = blockIdx.x;                 // one block per (b,c) row
    const float* row = x + (size_t)bc * N_;
    float acc = 0.f;
    for (int j = threadIdx.x; j < N_; j += blockDim.x) acc += row[j];
    __shared__ float sm[256];
    sm[threadIdx.x] = acc;
    __syncthreads();
    for (int off = 128; off > 0; off >>= 1) {
        if ((int)threadIdx.x < off) sm[threadIdx.x] += sm[threadIdx.x + off];
        __syncthreads();
    }
    if (threadIdx.x == 0) xs[bc] = sm[0];
}

// ---------------- Pass 2: fold weights (tiny, one block per batch) ----------------
// tsum[k] = theta_w[k,:].xs + N*theta_b[k]
// w[c]    = sum_k tsum[k]*phi_w[k,c] ;  s = sum_k tsum[k]*phi_b[k]
__global__ void nlb_fold(const float* __restrict__ xs,
                         const float* __restrict__ theta_w,
                         const float* __restrict__ theta_b,
                         const float* __restrict__ phi_w,
                         const float* __restrict__ phi_b,
                         float* __restrict__ wv, float* __restrict__ sv) {
    const int b = blockIdx.x;
    const int t = threadIdx.x;                 // 256 threads
    __shared__ float xs_l[C_];
    __shared__ float ts[CI_];
    __shared__ float sb[CI_];
    xs_l[t] = xs[b * C_ + t];
    __syncthreads();
    if (t < CI_) {
        const float* tw = theta_w + t * C_;
        float acc = 0.f;
        for (int c = 0; c < C_; ++c) acc += tw[c] * xs_l[c];
        ts[t] = acc + (float)N_ * theta_b[t];
    }
    __syncthreads();
    {
        float acc = 0.f;
        for (int k = 0; k < CI_; ++k) acc += ts[k] * phi_w[k * C_ + t];
        wv[b * C_ + t] = acc;
    }
    if (t < CI_) sb[t] = ts[t] * phi_b[t];
    __syncthreads();
    if (t == 0) {
        float acc = 0.f;
        for (int k = 0; k < CI_; ++k) acc += sb[k];
        sv[b] = acc;
    }
}

// ---------------- Pass 3: conf[b,j] = sigmoid((w.x[:,j] + s)/N^2) via fp32 WMMA ----------------
// One wave per 16-column tile. A = w chunk replicated across the 16 M-rows,
// B = 4x16 tile of x columns; result taken from D row M=0 (VGPR0, lanes 0..15).
__global__ void nlb_conf_wmma(const float* __restrict__ x,
                              const float* __restrict__ wv,
                              const float* __restrict__ sv,
                              float* __restrict__ conf) {
    const int lane = threadIdx.x & 31;
    const int wave = blockIdx.x * (blockDim.x >> 5) + (threadIdx.x >> 5);
    const int tilesPerBatch = N_ / 16;         // 256
    const int b  = wave / tilesPerBatch;
    const int j0 = (wave - b * tilesPerBatch) * 16;

    const float* xb = x  + (size_t)b * C_ * N_;
    const float* wb = wv + b * C_;

    const int n  = lane & 15;                  // N index within tile
    const int k0 = (lane >> 4) << 1;           // lanes 0-15 -> K=0,1 ; lanes 16-31 -> K=2,3

    v8f acc = {};
    for (int c0 = 0; c0 < C_; c0 += 4) {
        v2f a, bm;
        a.x  = wb[c0 + k0];                    // A[m][k] = w[c0+k]  (all rows identical)
        a.y  = wb[c0 + k0 + 1];
        bm.x = xb[(size_t)(c0 + k0)     * N_ + j0 + n];   // B[k][n] = x[c0+k, j0+n]
        bm.y = xb[(size_t)(c0 + k0 + 1) * N_ + j0 + n];
        acc = __builtin_amdgcn_wmma_f32_16x16x4_f32(
                  /*neg_a=*/false, a, /*neg_b=*/false, bm,
                  /*c_mod=*/(short)0, acc, /*reuse_a=*/false, /*reuse_b=*/false);
    }

    if (lane < 16) {                           // D[M=0][N=lane] lives in acc[0], lanes 0..15
        const float y = acc[0];
        const float m = (y + sv[b]) * (1.0f / ((float)N_ * (float)N_));
        conf[b * N_ + j0 + lane] = 1.0f / (1.0f + expf(-m));
    }
}

// ---------------- Pass 4: out = conf * x (float4 streaming) ----------------
__global__ void nlb_scale(const float* __restrict__ x,
                          const float* __restrict__ conf,
                          float* __restrict__ out) {
    const size_t i4 = ((size_t)blockIdx.x * blockDim.x + threadIdx.x) * 4;
    if (i4 >= (size_t)B_ * C_ * N_) return;
    const int b  = (int)(i4 >> 20);            // C_*N_ = 2^20
    const int nn = (int)(i4 & (size_t)(N_ - 1));
    const float4 xv = *(const float4*)(x + i4);
    const float* cb = conf + b * N_ + nn;      // nn is 4-aligned, conf contiguous in n
    float4 o;
    o.x = xv.x * cb[0];
    o.y = xv.y * cb[1];
    o.z = xv.z * cb[2];
    o.w = xv.w * cb[3];
    *(float4*)(out + i4) = o;
}

extern "C" void kernel_launch(void* const* d_in, const int* in_sizes, int n_in,
                              void* d_out, int out_size, void* d_ws, size_t ws_size,
                              hipStream_t stream) {
    const float* x       = (const float*)d_in[0];
    const float* theta_w = (const float*)d_in[1];
    const float* theta_b = (const float*)d_in[2];
    const float* phi_w   = (const float*)d_in[3];
    const float* phi_b   = (const float*)d_in[4];
    float* out = (float*)d_out;

    float* ws   = (float*)d_ws;
    float* xs   = ws + WS_XS;
    float* wv   = ws + WS_W;
    float* sv   = ws + WS_S;
    float* conf = ws + WS_CONF;

    // Pass 1: per-(b,c) row sums of x
    nlb_colsum<<<B_ * C_, 256, 0, stream>>>(x, xs);
    // Pass 2: fold theta/phi into per-batch w[c], s
    nlb_fold<<<B_, 256, 0, stream>>>(xs, theta_w, theta_b, phi_w, phi_b, wv, sv);
    // Pass 3: WMMA GEMV -> confidence map (1024 waves, 4 waves/block)
    nlb_conf_wmma<<<(B_ * (N_ / 16)) / 4, 128, 0, stream>>>(x, wv, sv, conf);
    // Pass 4: scale x by per-pixel confidence
    const size_t total4 = (size_t)B_ * C_ * N_ / 4;
    nlb_scale<<<(int)((total4 + 255) / 256), 256, 0, stream>>>(x, conf, out);
}